// AlignTTSLoss_83648783057106
// MI455X (gfx1250) — compile-verified
//
#include <hip/hip_runtime.h>
#include <hip/hip_bf16.h>
#include <math.h>

// ---------------------------------------------------------------------------
// AlignTTS MDN loss for gfx1250 (MI455X).
//   B=16, txt_T=320, mel_T=2000, n_mel=80  (K = 2*n_mel = 160)
// log_prob is a batched f32 GEMM (K=160) -> V_WMMA_F32_16X16X4_F32, split
// into two interleaved accumulators (x^2-half and x-half) so each melspec
// element is loaded once and the WMMA dependency chain is halved.
// Forward DP is sequential over mel_T; one 320-thread block per batch.
// ---------------------------------------------------------------------------

#define MDN_B     16
#define MDN_TXT   320
#define MDN_MEL   2000
#define MDN_NMEL  80
#define MDN_K     160
#define MDN_PAD   1000000000000.0f
#define MDN_EPS   1e-7f

typedef float v2f __attribute__((ext_vector_type(2)));
typedef float v8f __attribute__((ext_vector_type(8)));

// ---------------------------------------------------------------------------
// Prep: Abuf[b][j][c]      = exp(-logvar)                 (pairs with x^2)
//       Abuf[b][j][80+c]   = -2 * mu * exp(-logvar)       (pairs with x)
//       mm[b][j]           = sum_c mu^2*exp(-logvar) + logvar
// ---------------------------------------------------------------------------
__global__ void mdn_prep(const float* __restrict__ mu_logvar,
                         float* __restrict__ Abuf,
                         float* __restrict__ mm) {
  int idx = blockIdx.x * blockDim.x + threadIdx.x;   // over B*txt_T
  if (idx >= MDN_B * MDN_TXT) return;
  const float* row  = mu_logvar + (size_t)idx * MDN_K;
  float*       arow = Abuf      + (size_t)idx * MDN_K;
  float acc = 0.0f;
  for (int c = 0; c < MDN_NMEL; ++c) {
    float mu  = row[c];
    float lv  = row[MDN_NMEL + c];
    float inv = expf(-lv);
    arow[c]            = inv;
    arow[MDN_NMEL + c] = -2.0f * mu * inv;
    acc += mu * mu * inv + lv;
  }
  mm[idx] = acc;
}

// ---------------------------------------------------------------------------
// log_prob GEMM: one wave computes a 16(j) x 16(t) tile.
// 2 x 20 interleaved V_WMMA_F32_16X16X4_F32 over the two K-halves.
//
// A frag (16x4 f32, 2 VGPRs): lanes 0-15 -> K=k0,k0+1 ; lanes 16-31 -> K=k0+2,k0+3
// B frag (4x16 f32, 2 VGPRs): same K split, col = lane&15; built from one
//   melspec load per element: b0 = x^2 (pairs inv_var), b1 = x (pairs -2*mu*inv_var).
// C/D (16x16 f32, 8 VGPRs): VGPR v -> row v + 8*(lane>=16), col = lane&15.
// ---------------------------------------------------------------------------
__global__ __launch_bounds__(32) void mdn_logprob_wmma(
    const float* __restrict__ Abuf,     // [B][txt_T][160]
    const float* __restrict__ mm,       // [B][txt_T]
    const float* __restrict__ mel,      // [B][80][mel_T]
    float* __restrict__ logp) {         // [B][txt_T][mel_T]  (= d_out+1)
  const int tt   = blockIdx.x;          // 0..124  mel tile
  const int jt   = blockIdx.y;          // 0..19   txt tile
  const int b    = blockIdx.z;          // 0..15
  const int lane = threadIdx.x;         // 0..31
  const int half = lane >> 4;           // 0 or 1
  const int lm   = lane & 15;

  const int j0 = jt * 16;
  const int t0 = tt * 16;

  const float* Arow = Abuf + (size_t)(b * MDN_TXT + j0 + lm) * MDN_K;  // row = lm
  const float* melb = mel  + (size_t)b * MDN_NMEL * MDN_MEL + t0 + lm; // col = lm

  v8f acc0 = {};   // inv_var . x^2 half
  v8f acc1 = {};   // (-2 mu inv_var) . x half
  for (int k0 = 0; k0 < MDN_NMEL; k0 += 4) {
    const int ka = k0 + half * 2;
    // A fragments for both K-halves (8B-aligned v2f loads).
    v2f a0 = *(const v2f*)(Arow + ka);
    v2f a1 = *(const v2f*)(Arow + MDN_NMEL + ka);
    // One melspec load per element feeds both B fragments.
    float x0 = melb[(size_t)ka * MDN_MEL];
    float x1 = melb[(size_t)(ka + 1) * MDN_MEL];
    v2f b0; b0.x = x0 * x0; b0.y = x1 * x1;
    v2f b1; b1.x = x0;      b1.y = x1;
    acc0 = __builtin_amdgcn_wmma_f32_16x16x4_f32(
        false, a0, false, b0, (short)0, acc0, false, false);
    acc1 = __builtin_amdgcn_wmma_f32_16x16x4_f32(
        false, a1, false, b1, (short)0, acc1, false, false);
  }

  // Combine halves, apply row bias mm and -1/160 scale, store tile.
  const float* mmb  = mm   + b * MDN_TXT + j0 + half * 8;
  float*       outb = logp + (size_t)(b * MDN_TXT + j0 + half * 8) * MDN_MEL
                           + t0 + lm;
#pragma unroll
  for (int v = 0; v < 8; ++v) {
    float r = -(1.0f / (float)MDN_K) * (acc0[v] + acc1[v] + mmb[v]);
    outb[(size_t)v * MDN_MEL] = r;
  }
}

// ---------------------------------------------------------------------------
// Forward DP: one block per batch, thread j owns txt position j.
// alpha (shifted by 1, alpha[0] = -PAD sentinel) lives in LDS.
// tail[j] = logsumexp(alpha[j+1]+EPS, alpha[j]+EPS) + log_prob[b,j,t]
// Capture alpha at (text_len-1, mel_len-1), divide by mel_len.
// ---------------------------------------------------------------------------
__global__ __launch_bounds__(MDN_TXT) void mdn_dp(
    const float* __restrict__ logp,     // [B][txt_T][mel_T]
    const int* __restrict__ text_len,
    const int* __restrict__ mel_len,
    float* __restrict__ partial) {      // [B]
  const int b = blockIdx.x;
  const int j = threadIdx.x;            // 0..319
  __shared__ float alpha[MDN_TXT + 1];

  const int tl = text_len[b];
  const int ml = mel_len[b];
  const float* lprow = logp + (size_t)b * MDN_TXT * MDN_MEL
                            + (size_t)j * MDN_MEL;

  if (j == 0) alpha[0] = -MDN_PAD;
  alpha[j + 1] = (j == 0) ? lprow[0] : -MDN_PAD;
  __syncthreads();

  const bool capture_me = (j == tl - 1);
  float cap = 0.0f;
  if (capture_me && ml == 1) cap = alpha[j + 1];

  for (int t = 1; t < MDN_MEL; ++t) {
    // Prefetch ahead on this lane's strided row (global_prefetch_b8 path).
    if ((t & 15) == 1) __builtin_prefetch(lprow + t + 16, 0, 1);

    float a_hi = alpha[j + 1];          // stay at same txt position
    float a_lo = alpha[j];              // advance from previous txt position
    float mx = fmaxf(a_hi, a_lo);
    float mn = fminf(a_hi, a_lo);
    float tail = mx + log1pf(expf(mn - mx)) + MDN_EPS + lprow[t];
    __syncthreads();
    alpha[j + 1] = tail;
    __syncthreads();
    if (capture_me && t == ml - 1) cap = tail;
  }
  if (capture_me) partial[b] = cap / (float)ml;
}

__global__ void mdn_finalize(const float* __restrict__ partial,
                             float* __restrict__ out) {
  if (blockIdx.x == 0 && threadIdx.x == 0) {
    float s = 0.0f;
    for (int b = 0; b < MDN_B; ++b) s += partial[b];
    out[0] = -s * (1.0f / (float)MDN_B);
  }
}

// ---------------------------------------------------------------------------
// Launch. d_out layout: [0] = mdn_loss, [1..] = log_prob flat (16*320*2000).
// Workspace: Abuf (3.28 MB) + mm (20 KB) + partial (64 B).
// ---------------------------------------------------------------------------
extern "C" void kernel_launch(void* const* d_in, const int* in_sizes, int n_in,
                              void* d_out, int out_size, void* d_ws, size_t ws_size,
                              hipStream_t stream) {
  const float* mu_logvar = (const float*)d_in[0];   // [16,320,160]
  const float* melspec   = (const float*)d_in[1];   // [16,80,2000]
  const int*   text_len  = (const int*)d_in[2];     // [16]
  const int*   mel_len   = (const int*)d_in[3];     // [16]

  float* out  = (float*)d_out;
  float* logp = out + 1;

  float* Abuf    = (float*)d_ws;                           // 16*320*160
  float* mm      = Abuf + (size_t)MDN_B * MDN_TXT * MDN_K; // 16*320
  float* partial = mm + MDN_B * MDN_TXT;                   // 16

  {
    int rows = MDN_B * MDN_TXT;
    mdn_prep<<<(rows + 255) / 256, 256, 0, stream>>>(mu_logvar, Abuf, mm);
  }
  {
    dim3 grid(MDN_MEL / 16, MDN_TXT / 16, MDN_B);   // 125 x 20 x 16
    mdn_logprob_wmma<<<grid, 32, 0, stream>>>(Abuf, mm, melspec, logp);
  }
  mdn_dp<<<MDN_B, MDN_TXT, 0, stream>>>(logp, text_len, mel_len, partial);
  mdn_finalize<<<1, 1, 0, stream>>>(partial, out);
}